// SkipGram_68496138437105
// MI455X (gfx1250) — compile-verified
//
#include <hip/hip_runtime.h>
#include <hip/hip_bf16.h>

typedef __attribute__((ext_vector_type(2))) float v2f;
typedef __attribute__((ext_vector_type(8))) float v8f;

#define V_ROWS   100000
#define D_DIM    1000
#define P_PAIRS  131072
#define N_SLICES 64
#define ROWS_PER_SLICE 1563   // ceil(100000/64)
#define PAIR_BLOCKS 1024      // 1024 blocks * 8 waves * 16 pairs = 131072

// ---------------------------------------------------------------------------
// Kernel 1a: partial column sums. grid = (4, 64), block = 256.
// partial[slice*1024 + c] = sum of E[r][c] for r in slice.
// ---------------------------------------------------------------------------
__global__ void colsum_partial_kernel(const float* __restrict__ E,
                                      float* __restrict__ partial) {
    int c     = blockIdx.x * 256 + threadIdx.x;   // 0..1023
    int slice = blockIdx.y;                       // 0..63
    int r0 = slice * ROWS_PER_SLICE;
    int r1 = r0 + ROWS_PER_SLICE;
    if (r1 > V_ROWS) r1 = V_ROWS;
    float s = 0.0f;
    if (c < D_DIM) {
        const float* p = E + (size_t)r0 * D_DIM + c;
        #pragma unroll 4
        for (int r = r0; r < r1; ++r) {
            s += *p;
            p += D_DIM;
        }
    }
    partial[slice * 1024 + c] = s;   // write all 1024 columns (0 for pad)
}

// ---------------------------------------------------------------------------
// Kernel 1b: reduce slice partials (fixed order -> deterministic).
// grid = 4, block = 256.
// ---------------------------------------------------------------------------
__global__ void colsum_final_kernel(const float* __restrict__ partial,
                                    float* __restrict__ colsum) {
    int c = blockIdx.x * 256 + threadIdx.x;       // 0..1023
    float s = 0.0f;
    #pragma unroll 8
    for (int k = 0; k < N_SLICES; ++k)
        s += partial[k * 1024 + c];
    colsum[c] = s;
}

// ---------------------------------------------------------------------------
// Kernel 2: per-pair cost via WMMA f32 16x16x4, fused pair-dot + norm-dot.
// One wave32 handles 16 pairs. grid = 1024, block = 256 (8 waves).
//
//   cd += A(E[i] rows 16x4) x B(E[j] cols 4x16)      -> diag = dot(E[i_m],E[j_m])
//   cn += A(E[i] rows 16x4) x B(col_sum bcast 4x16)  -> diag = dot(E[i_m],colsum)
//
// 32-bit A 16x4 layout: lane l holds row M=(l&15), K = 2*(l>>4)..+1 (2 VGPRs).
// B 4x16 mirrors it (lane = column N). C/D diag: lanes 0-7 elem=lane,
// lanes 24-31 elem=lane-24.
// ---------------------------------------------------------------------------
__global__ void pair_cost_kernel(const float* __restrict__ E,
                                 const int*  __restrict__ iv,
                                 const int*  __restrict__ jv,
                                 const float* __restrict__ colsum,
                                 float* __restrict__ block_partials) {
    __shared__ float lds_cs[D_DIM];
    __shared__ float wave_sums[8];

    int tid = threadIdx.x;
    // Stage col_sum into LDS once per block.
    for (int c = tid; c < D_DIM; c += 256) lds_cs[c] = colsum[c];
    __syncthreads();

    int wave = tid >> 5;
    int lane = tid & 31;
    int pair_base = (blockIdx.x * 8 + wave) * 16;
    int m  = lane & 15;                 // pair within wave
    int kh = (lane >> 4) << 1;          // K sub-offset: 0 or 2

    int rowI = iv[pair_base + m];
    int rowJ = jv[pair_base + m];
    const float* Ei = E + (size_t)rowI * D_DIM;
    const float* Ej = E + (size_t)rowJ * D_DIM;

    v8f cd = {};
    v8f cn = {};
    #pragma unroll 2
    for (int k = 0; k < D_DIM; k += 4) {
        int ko = k + kh;
        v2f a = *(const v2f*)(Ei + ko);          // A: rows of E[i]
        v2f b = *(const v2f*)(Ej + ko);          // B: cols of E[j]
        v2f s;                                   // B: colsum broadcast
        s.x = lds_cs[ko];
        s.y = lds_cs[ko + 1];
        cd = __builtin_amdgcn_wmma_f32_16x16x4_f32(false, a, false, b,
                                                   (short)0, cd, false, false);
        cn = __builtin_amdgcn_wmma_f32_16x16x4_f32(false, a, false, s,
                                                   (short)0, cn, false, false);
    }

    // Extract diagonal: lanes 0-7 -> elem lane; lanes 24-31 -> elem lane-24.
    bool active = (lane < 8) || (lane >= 24);
    int  r = lane & 7;
    float dsel, nsel;
    dsel = cd[0]; dsel = (r == 1) ? cd[1] : dsel; dsel = (r == 2) ? cd[2] : dsel;
    dsel = (r == 3) ? cd[3] : dsel; dsel = (r == 4) ? cd[4] : dsel;
    dsel = (r == 5) ? cd[5] : dsel; dsel = (r == 6) ? cd[6] : dsel;
    dsel = (r == 7) ? cd[7] : dsel;
    nsel = cn[0]; nsel = (r == 1) ? cn[1] : nsel; nsel = (r == 2) ? cn[2] : nsel;
    nsel = (r == 3) ? cn[3] : nsel; nsel = (r == 4) ? cn[4] : nsel;
    nsel = (r == 5) ? cn[5] : nsel; nsel = (r == 6) ? cn[6] : nsel;
    nsel = (r == 7) ? cn[7] : nsel;

    float cost = 0.0f;
    if (active) {
        // -log(exp(d)/exp(n) + 1e-8) == -log(exp(d - n) + 1e-8)
        cost = -__logf(__expf(dsel - nsel) + 1e-8f);
    }

    // Deterministic wave reduction (fixed shuffle tree).
    #pragma unroll
    for (int off = 16; off > 0; off >>= 1)
        cost += __shfl_down(cost, off, 32);
    if (lane == 0) wave_sums[wave] = cost;
    __syncthreads();
    if (tid == 0) {
        float s = 0.0f;
        #pragma unroll
        for (int w = 0; w < 8; ++w) s += wave_sums[w];
        block_partials[blockIdx.x] = s;
    }
}

// ---------------------------------------------------------------------------
// Kernel 3: finalize. 1 block of 32 threads, fixed-order reduce -> mean.
// ---------------------------------------------------------------------------
__global__ void finalize_kernel(const float* __restrict__ block_partials,
                                float* __restrict__ out) {
    int lane = threadIdx.x;
    float s = 0.0f;
    for (int k = lane; k < PAIR_BLOCKS; k += 32)
        s += block_partials[k];
    #pragma unroll
    for (int off = 16; off > 0; off >>= 1)
        s += __shfl_down(s, off, 32);
    if (lane == 0) out[0] = s / (float)P_PAIRS;
}

// ---------------------------------------------------------------------------
extern "C" void kernel_launch(void* const* d_in, const int* in_sizes, int n_in,
                              void* d_out, int out_size, void* d_ws, size_t ws_size,
                              hipStream_t stream) {
    const float* E  = (const float*)d_in[0];   // [V, D] fp32
    const int*   iv = (const int*)d_in[1];     // [P] int32
    const int*   jv = (const int*)d_in[2];     // [P] int32
    float* out = (float*)d_out;                // scalar fp32

    // Workspace layout (floats): [0,1024) colsum | [1024, 1024+64*1024) slice
    // partials | then 1024 block partials.  Total ~270 KB.
    float* ws      = (float*)d_ws;
    float* colsum  = ws;
    float* partial = ws + 1024;
    float* bpart   = ws + 1024 + N_SLICES * 1024;

    colsum_partial_kernel<<<dim3(4, N_SLICES), 256, 0, stream>>>(E, partial);
    colsum_final_kernel<<<4, 256, 0, stream>>>(partial, colsum);
    pair_cost_kernel<<<PAIR_BLOCKS, 256, 0, stream>>>(E, iv, jv, colsum, bpart);
    finalize_kernel<<<1, 32, 0, stream>>>(bpart, out);
}